// ActivationFilter_76931454206624
// MI455X (gfx1250) — compile-verified
//
#include <hip/hip_runtime.h>
#include <hip/hip_bf16.h>

typedef float v2f __attribute__((ext_vector_type(2)));
typedef float v8f __attribute__((ext_vector_type(8)));

#define NB 8
#define HH 128
#define WW 128
#define CC 128
#define NEG_SLOPE 0.01f

// Keys cubic, a = -0.5 (jax.image 'cubic'/'bicubic' kernel)
__device__ __forceinline__ float keys_cubic(float x) {
  x = fabsf(x);
  if (x >= 2.0f) return 0.0f;
  if (x < 1.0f) return (1.5f * x - 2.5f) * x * x + 1.0f;
  return ((-0.5f * x + 2.5f) * x - 4.0f) * x + 2.0f;
}

// One bicubic-upsample weight row: output = fine row fg (global), taps over a
// 16-wide coarse window starting at `base`.  OOB taps dropped + renormalized
// (matches jax.image.resize edge handling).
__device__ __forceinline__ void fill_up_row(float* row, int fg, int base, int N) {
  float coord = (float)fg * 0.5f - 0.25f;  // (f+0.5)/2 - 0.5
  float wv[16];
  float s = 0.0f;
#pragma unroll
  for (int k = 0; k < 16; ++k) {
    int gi = base + k;
    float wt = (gi >= 0 && gi < N) ? keys_cubic((float)gi - coord) : 0.0f;
    wv[k] = wt;
    s += wt;
  }
  float inv = (s > 0.0f) ? 1.0f / s : 0.0f;
#pragma unroll
  for (int k = 0; k < 16; ++k) row[k] = wv[k] * inv;
}

// One antialiased bicubic-downsample weight row: output = coarse row og
// (global), taps over a 24-wide fine window starting at `base`.  Kernel support
// scaled by 2 (antialias at scale 1/2); scale constant absorbed by renorm.
__device__ __forceinline__ void fill_dn_row(float* row, int og, int base, int N) {
  float coord = 2.0f * (float)og + 0.5f;  // (o+0.5)*2 - 0.5
  float wv[24];
  float s = 0.0f;
#pragma unroll
  for (int k = 0; k < 24; ++k) {
    int gi = base + k;
    float wt = (gi >= 0 && gi < N) ? keys_cubic(((float)gi - coord) * 0.5f) : 0.0f;
    wv[k] = wt;
    s += wt;
  }
  float inv = (s > 0.0f) ? 1.0f / s : 0.0f;
#pragma unroll
  for (int k = 0; k < 24; ++k) row[k] = wv[k] * inv;
}

// D(16x16) = Wm[m0..m0+16, 0..K) x B(K x 16), K in {16,24}, chunks of K=4 via
// V_WMMA_F32_16X16X4_F32.  B element (k,n) lives at Bm[bBase + k*bStride + n].
// Layouts per CDNA5 ISA 7.12.2:
//   A 16x4: lanes 0-15 hold K={0,1}, lanes 16-31 hold K={2,3}, M = lane%16
//   B 4x16: lanes 0-15 hold K={0,1}, lanes 16-31 hold K={2,3}, N = lane%16
//   C/D:    vgpr r holds M = r + 8*(lane>=16), N = lane%16
template <int K>
__device__ __forceinline__ v8f band_gemm(const float* Wm, int wK, int m0,
                                         const float* Bm, int bBase, int bStride,
                                         int lane) {
  const int half = lane >> 4;
  const int mn = lane & 15;
  v8f acc = {};
#pragma unroll
  for (int kb = 0; kb < K; kb += 4) {
    v2f a, bv;
    const float* wr = Wm + (m0 + mn) * wK + kb + 2 * half;
    a.x = wr[0];
    a.y = wr[1];
    const float* bp = Bm + bBase + (kb + 2 * half) * bStride + mn;
    bv.x = bp[0];
    bv.y = bp[bStride];
    acc = __builtin_amdgcn_wmma_f32_16x16x4_f32(false, a, false, bv, 0, acc,
                                                false, false);
  }
  return acc;
}

// Fully fused: coarse(16x16x16ch in LDS) --H-up--> fineH(24..32 x16x16)
// --W-up+leaky--> fineHW(24x24..32x16) --W-down--> wdown(24x8..16x16)
// --H-down--> 8x8x16 output tile.  One HBM read + one HBM write per element.
__global__ __launch_bounds__(256) void af_fused_kernel(const float* __restrict__ x,
                                                       float* __restrict__ out) {
  __shared__ float sCoarse[16 * 16 * 16];  // [h16][w16][c16]
  __shared__ float sFineH[32 * 16 * 16];   // [fh32][w16][c16]
  __shared__ float sFineHW[24 * 32 * 16];  // [fh24][fw32][c16]
  __shared__ float sWdown[24 * 16 * 16];   // [fh24][wc16][c16]
  __shared__ float wUpH[32 * 16];
  __shared__ float wUpW[32 * 16];
  __shared__ float wDnW[16 * 24];
  __shared__ float wDnH[16 * 24];

  const int tid = threadIdx.x;
  const int lane = tid & 31;
  const int wave = tid >> 5;
  const int half = lane >> 4;
  const int n = lane & 15;

  const int tw = blockIdx.x & 15;   // 16 w-tiles of 8
  const int th = blockIdx.x >> 4;   // 16 h-tiles of 8
  const int cb = blockIdx.y << 4;   // 8 channel groups of 16
  const int b = blockIdx.z;

  const int H0 = th * 8, W0 = tw * 8;          // coarse output tile origin
  const int C0h = H0 - 4, C0w = W0 - 4;        // coarse input window origin (16)
  const int F0h = th * 16 - 4, F0w = tw * 16 - 4;  // fine window origin (24 used)

  // ---- stage 0a: async-DMA the 16x16x16 coarse window into LDS ----
  // GLOBAL_LOAD_ASYNC_TO_LDS_B128 copies 16B chunks memory->LDS without
  // touching VGPRs (tracked by ASYNCcnt).  OOB (h,w) slots are zero-filled
  // with plain ds stores (disjoint addresses, so no ordering hazard).
  {
    const unsigned ldsBase = (unsigned)(uintptr_t)(void*)sCoarse;
    for (int i = tid; i < 16 * 16 * 4; i += 256) {  // 1024 x 16B chunks
      int wp = (i >> 2) & 15, hp = i >> 6;
      int gh = C0h + hp, gw = C0w + wp;
      if (gh >= 0 && gh < HH && gw >= 0 && gw < WW) {
        const float* gp =
            x + (((size_t)b * HH + gh) * WW + gw) * CC + cb + ((i & 3) << 2);
        unsigned lds = ldsBase + (unsigned)i * 16u;
        asm volatile("global_load_async_to_lds_b128 %0, %1, off"
                     :
                     : "v"(lds), "v"(gp)
                     : "memory");
      } else {
        float4 z = make_float4(0.f, 0.f, 0.f, 0.f);
        *reinterpret_cast<float4*>(&sCoarse[i * 4]) = z;
      }
    }
  }

  // ---- stage 0b: build the four banded filter matrices (overlaps the DMA) --
  if (tid < 32) {
    fill_up_row(wUpH + tid * 16, F0h + tid, C0h, HH);
  } else if (tid < 64) {
    int r = tid - 32;
    fill_up_row(wUpW + r * 16, F0w + r, C0w, WW);
  } else if (tid < 80) {
    int r = tid - 64;
    fill_dn_row(wDnW + r * 24, W0 + r, F0w, 2 * WW);
  } else if (tid < 96) {
    int r = tid - 80;
    fill_dn_row(wDnH + r * 24, H0 + r, F0h, 2 * HH);
  }

  // drain this wave's async LDS writes, then rendezvous the workgroup
  asm volatile("s_wait_asynccnt 0" ::: "memory");
  __syncthreads();

  // ---- stage 1: H-upsample.  Per coarse-w column: (32 fine-h x 16c) ----
  for (int job = wave; job < 32; job += 8) {
    int wp = job & 15, m0 = (job >> 4) * 16;
    v8f d = band_gemm<16>(wUpH, 16, m0, sCoarse, wp * 16, 256, lane);
#pragma unroll
    for (int r = 0; r < 8; ++r) {
      int fh = m0 + r + 8 * half;
      sFineH[(fh * 16 + wp) * 16 + n] = d[r];
    }
  }
  __syncthreads();

  // ---- stage 2: W-upsample + leaky ReLU.  Per fine-h row (24 used) ----
  for (int job = wave; job < 48; job += 8) {
    int fh = job % 24, m0 = (job / 24) * 16;
    v8f d = band_gemm<16>(wUpW, 16, m0, sFineH, fh * 256, 16, lane);
#pragma unroll
    for (int r = 0; r < 8; ++r) {
      int fw = m0 + r + 8 * half;
      float v = d[r];
      v = (v > 0.0f) ? v : NEG_SLOPE * v;
      sFineHW[(fh * 32 + fw) * 16 + n] = v;
    }
  }
  __syncthreads();

  // ---- stage 3: W-downsample (K=24 fine taps -> 16 coarse w, 8 valid) ----
  for (int fh = wave; fh < 24; fh += 8) {
    v8f d = band_gemm<24>(wDnW, 24, 0, sFineHW, fh * 512, 16, lane);
#pragma unroll
    for (int r = 0; r < 8; ++r) {
      int wc = r + 8 * half;
      sWdown[(fh * 16 + wc) * 16 + n] = d[r];
    }
  }
  __syncthreads();

  // ---- stage 4: H-downsample + store the 8x8x16 output tile ----
  for (int wp = wave; wp < 8; wp += 8) {
    v8f d = band_gemm<24>(wDnH, 24, 0, sWdown, wp * 16, 256, lane);
    if (half == 0) {  // rows 0..7 are the valid coarse-h outputs
#pragma unroll
      for (int r = 0; r < 8; ++r) {
        out[(((size_t)b * HH + (H0 + r)) * WW + (W0 + wp)) * CC + cb + n] = d[r];
      }
    }
  }
}

extern "C" void kernel_launch(void* const* d_in, const int* in_sizes, int n_in,
                              void* d_out, int out_size, void* d_ws, size_t ws_size,
                              hipStream_t stream) {
  const float* x = (const float*)d_in[0];
  float* out = (float*)d_out;
  dim3 grid(256, 8, 8);  // (h-tile x w-tile), channel-group, batch
  af_fused_kernel<<<grid, dim3(256), 0, stream>>>(x, out);
}